// papm_diffusive_flows_56315611185396
// MI455X (gfx1250) — compile-verified
//
#include <hip/hip_runtime.h>
#include <hip/hip_bf16.h>

typedef __attribute__((ext_vector_type(2))) float v2f;
typedef __attribute__((ext_vector_type(8))) float v8f;

// Problem geometry
#define HH 512
#define WW 512
#define TROWS 16     // output rows per workgroup tile
#define TCOLS 128    // output cols per workgroup tile (8 waves x 16 cols)
#define LROWS 20     // staged rows  (16 + 4 halo)
#define LCOLS 132    // staged cols  (128 + 4 halo)
#define LPITCH 136   // LDS row pitch (even -> 8B-aligned v2f loads; spreads banks)

__global__ __launch_bounds__(256, 4)
void papm_diffusive_flows_kernel(const float* __restrict__ x,
                                 const float* __restrict__ Wu,
                                 const float* __restrict__ bu,
                                 const float* __restrict__ Wv,
                                 const float* __restrict__ bv,
                                 float* __restrict__ out)
{
    __shared__ __align__(16) float tile[LROWS * LPITCH];
    __shared__ float wlds[32];

    const int t  = threadIdx.x;
    const int wg = blockIdx.x;
    // grid = 128 batches * 2 channels * 32 row-tiles * 4 col-strips = 32768
    const int tc = wg & 3;
    const int tr = (wg >> 2) & 31;
    const int ch = (wg >> 7) & 1;
    const int b  = wg >> 8;

    const int r0 = tr * TROWS;
    const int c0 = tc * TCOLS;

    const float* __restrict__ Wsel = ch ? Wv : Wu;
    const float  bias  = ch ? bv[0] : bu[0];
    const float  scale = ch ? 0.005f : 0.001f;   // DV : DU

    if (t < 25) wlds[t] = Wsel[t];

    const float* __restrict__ plane =
        x + (size_t)(b * 2 + ch) * (size_t)(HH * WW);

    // ---- Stage 20x132 wrapped halo region into LDS (wrap via &511) ----
    for (int idx = t; idx < LROWS * LCOLS; idx += 256) {
        int i  = idx / LCOLS;
        int j  = idx - i * LCOLS;
        int gr = (r0 + i - 2) & (HH - 1);
        int gc = (c0 + j - 2) & (WW - 1);
        tile[i * LPITCH + j] = plane[gr * WW + gc];
    }
    __syncthreads();

    // ---- Per-wave 16x16 output subtile via 25x V_WMMA_F32_16X16X4_F32 ----
    const int lane = t & 31;
    const int wv   = t >> 5;            // wave id 0..7 -> col offset 16*wv
    const int n    = lane & 15;         // N for B/C/D, M for A
    const int kk   = (lane >> 4) << 1;  // K base: 0 (lanes 0-15) or 2 (16-31)

    // Precompute banded-weight B fragments: B[k][n] = W[dy][4j+k-n] (0<=d<=4)
    v2f bfrag[25];
#pragma unroll
    for (int dy = 0; dy < 5; ++dy) {
#pragma unroll
        for (int j = 0; j < 5; ++j) {
            const int d0 = 4 * j + kk + 0 - n;
            const int d1 = 4 * j + kk + 1 - n;
            v2f bb;
            bb.x = (d0 >= 0 && d0 <= 4) ? wlds[dy * 5 + d0] : 0.0f;
            bb.y = (d1 >= 0 && d1 <= 4) ? wlds[dy * 5 + d1] : 0.0f;
            bfrag[dy * 5 + j] = bb;
        }
    }

    v8f acc = {0.f, 0.f, 0.f, 0.f, 0.f, 0.f, 0.f, 0.f};
    const int colbase = wv * 16;

#pragma unroll
    for (int dy = 0; dy < 5; ++dy) {
#pragma unroll
        for (int j = 0; j < 5; ++j) {
            // A[m][k] = tile[(m+dy)][colbase + 4j + k]; lane holds m=n, k=kk+v
            const float* ap = &tile[(n + dy) * LPITCH + colbase + 4 * j + kk];
            v2f a = *(const v2f*)ap;    // 8B-aligned ds_load_b64
            acc = __builtin_amdgcn_wmma_f32_16x16x4_f32(
                false, a, false, bfrag[dy * 5 + j],
                (short)0, acc, false, false);
        }
    }

    // ---- Scale + bias, store 16x16 tile (C/D layout: M = i + 8*(lane/16)) ----
    const float add = scale * bias;
    float* __restrict__ oplane =
        out + (size_t)(b * 2 + ch) * (size_t)(HH * WW);
    const int rowhi = (lane >> 4) << 3;
    const int ocol  = c0 + colbase + n;
#pragma unroll
    for (int i = 0; i < 8; ++i) {
        const int orow = r0 + rowhi + i;
        oplane[orow * WW + ocol] = scale * acc[i] + add;
    }
}

extern "C" void kernel_launch(void* const* d_in, const int* in_sizes, int n_in,
                              void* d_out, int out_size, void* d_ws, size_t ws_size,
                              hipStream_t stream) {
    (void)in_sizes; (void)n_in; (void)out_size; (void)d_ws; (void)ws_size;
    const float* x  = (const float*)d_in[0];
    const float* Wu = (const float*)d_in[1];
    const float* bu = (const float*)d_in[2];
    const float* Wv = (const float*)d_in[3];
    const float* bv = (const float*)d_in[4];
    float* out = (float*)d_out;

    // 128 batches * 2 channels * (512/16) row-tiles * (512/128) col-strips
    dim3 grid(128 * 2 * 32 * 4);
    dim3 block(256);
    papm_diffusive_flows_kernel<<<grid, block, 0, stream>>>(x, Wu, bu, Wv, bv, out);
}